// TTConv_60473139528507
// MI455X (gfx1250) — compile-verified
//
#include <hip/hip_runtime.h>

typedef float v2f __attribute__((ext_vector_type(2)));
typedef float v8f __attribute__((ext_vector_type(8)));

#define MS   272   // 16x16 matrix slot, row pitch 17 floats (conflict-free column access)
#define XGP  544   // XG pitch per K-pair row: 512 data + 32 pad (bank stagger for hi half)
#define WLP  160   // WLp pitch per K-pair row: 128 data + 32 pad (bank stagger)

// Cayley transform of one 16x16 matrix, held column-wise across a 16-lane group.
// Lane c owns column c: a[r] = (I+S)[r][c], bm[r] = (I-S)[r][c].
// Gauss-Jordan without pivoting is valid: x^T(I+S)x = |x|^2 > 0.
__device__ __forceinline__ void cayley16(float (&a)[16], float (&bm)[16]) {
#pragma unroll
  for (int p = 0; p < 16; ++p) {
    float piv = __shfl(a[p], p, 16);   // A[p][p]
    float rpiv = 1.0f / piv;
    a[p] *= rpiv;
    bm[p] *= rpiv;
#pragma unroll
    for (int r = 0; r < 16; ++r) {
      if (r == p) continue;
      float f = __shfl(a[r], p, 16);   // A[r][p] (lane p's column)
      a[r]  = fmaf(-f, a[p],  a[r]);
      bm[r] = fmaf(-f, bm[p], bm[r]);
    }
  }
}

// Cayley-transform 64 matrices sitting in LDS (pitch-17 rows), 16 at a time.
__device__ __forceinline__ void cayley_lds_batch(float* mats, int tid) {
  int grp = tid >> 4, lc = tid & 15;
#pragma unroll 1
  for (int round = 0; round < 4; ++round) {
    float* base = mats + (round * 16 + grp) * MS;
    float a[16], bm[16];
#pragma unroll
    for (int r = 0; r < 16; ++r) {
      float d = base[r * 17 + lc] - base[lc * 17 + r];  // S = W - W^T
      float id = (r == lc) ? 1.0f : 0.0f;
      a[r] = id + d;
      bm[r] = id - d;
    }
    cayley16(a, bm);
#pragma unroll
    for (int r = 0; r < 16; ++r) base[r * 17 + lc] = bm[r];
  }
}

// Fused out1 path: cayley(core1) -> mix(w1,.) -> cayley(.) -> out1. One block per (b,s).
__global__ __launch_bounds__(256) void tt_out1_fused(
    const float* __restrict__ core1, const float* __restrict__ weight,
    float* __restrict__ out1) {
  extern __shared__ float smem[];
  float* XL  = smem;                 // 64*MS floats: matrices; later reused for GEMM output
  float* XG  = smem + 64 * MS;       // 32*XGP floats: K-pair-interleaved B operand
  float* WLp = XG + 32 * XGP;        // 32*WLP floats: K-pair-interleaved A operand

  int b = blockIdx.x >> 4;
  int s = blockIdx.x & 15;
  int tid = threadIdx.x;

  // core1[i,b,k,s,j]: strides i=1048576, b=4096, k=256, s=16, j=1
  // Vector (b128) global loads; scalar LDS stores keep the pitch-17 layout.
  const float* src = core1 + (size_t)b * 4096 + s * 16;
  for (int idx = tid; idx < 4096; idx += 256) {
    int q = idx & 3, k = (idx >> 2) & 15, i = idx >> 6;
    float4 v = *(const float4*)(src + (size_t)i * 1048576 + k * 256 + q * 4);
    float* d = &XL[i * MS + k * 17 + q * 4];
    d[0] = v.x; d[1] = v.y; d[2] = v.z; d[3] = v.w;
  }
  // w1[i,s,c] -> WLp[kp][c][par], i = 2*kp+par  (A[M=c][K=i])
  const float* wsrc = weight + (16 + s) * 64;
  for (int idx = tid; idx < 4096; idx += 256) {
    int i = idx >> 6, c = idx & 63;
    WLp[(i >> 1) * WLP + c * 2 + (i & 1)] = wsrc[i * 3072 + c];
  }
  __syncthreads();

  cayley_lds_batch(XL, tid);         // c1 for all 64 i-slices
  __syncthreads();

  // Repack: XG[kp][n][par] = c1[i=2kp+par][n], n = k*16+j  (enables ds_load_b64 B-frags)
  for (int idx = tid; idx < 16384; idx += 256) {
    int par = idx & 1, n = (idx >> 1) & 255, kp = idx >> 9;
    int i = kp * 2 + par;
    XG[kp * XGP + n * 2 + par] = XL[i * MS + (n >> 4) * 17 + (n & 15)];
  }
  __syncthreads();

  // GEMM: D[64 x 256] = A[64x64] * B[64x256]; output overwrites XL (dead after repack).
  int wave = tid >> 5, lane = tid & 31, l16 = lane & 15, hi = lane >> 4;
  int mt = wave >> 1;                // M-tile constant per wave -> A frags shared
  int ntb = (wave & 1) * 8;

  v2f a_frag[16];                    // all K-step A fragments, hoisted once
#pragma unroll
  for (int k4 = 0; k4 < 16; ++k4)
    a_frag[k4] = *(const v2f*)&WLp[(k4 * 2 + hi) * WLP + (mt * 16 + l16) * 2];

#pragma unroll 1
  for (int qp = 0; qp < 4; ++qp) {   // 2 N-tiles per iteration, 4 independent acc chains
    int nt0 = ntb + qp * 2, nt1 = nt0 + 1;
    const float* bp0 = &XG[hi * XGP + (nt0 * 16 + l16) * 2];
    const float* bp1 = &XG[hi * XGP + (nt1 * 16 + l16) * 2];
    v8f a00 = {}, a01 = {}, a10 = {}, a11 = {};
#pragma unroll
    for (int k2 = 0; k2 < 8; ++k2) {
      v2f b0e = *(const v2f*)&bp0[(k2 * 4) * XGP];
      v2f b1e = *(const v2f*)&bp1[(k2 * 4) * XGP];
      v2f b0o = *(const v2f*)&bp0[(k2 * 4 + 2) * XGP];
      v2f b1o = *(const v2f*)&bp1[(k2 * 4 + 2) * XGP];
      a00 = __builtin_amdgcn_wmma_f32_16x16x4_f32(false, a_frag[k2 * 2], false, b0e,
                                                  (short)0, a00, false, false);
      a10 = __builtin_amdgcn_wmma_f32_16x16x4_f32(false, a_frag[k2 * 2], false, b1e,
                                                  (short)0, a10, false, false);
      a01 = __builtin_amdgcn_wmma_f32_16x16x4_f32(false, a_frag[k2 * 2 + 1], false, b0o,
                                                  (short)0, a01, false, false);
      a11 = __builtin_amdgcn_wmma_f32_16x16x4_f32(false, a_frag[k2 * 2 + 1], false, b1o,
                                                  (short)0, a11, false, false);
    }
    v8f acc0 = a00 + a01;
    v8f acc1 = a10 + a11;
#pragma unroll
    for (int v = 0; v < 8; ++v) {
      int M = mt * 16 + v + hi * 8;  // C/D layout: VGPR v -> rows v, v+8
      XL[M * MS + nt0 * 17 + l16] = acc0[v];
      XL[M * MS + nt1 * 17 + l16] = acc1[v];
    }
  }
  __syncthreads();

  cayley_lds_batch(XL, tid);         // second cayley over (k,j) per channel c
  __syncthreads();

  // out1[c,b,k,s,j]: strides c=1048576, b=4096, k=256, s=16, j=1
  // Scalar LDS reads, vector (b128) global stores.
  float* dst = out1 + (size_t)b * 4096 + s * 16;
  for (int idx = tid; idx < 4096; idx += 256) {
    int q = idx & 3, k = (idx >> 2) & 15, c = idx >> 6;
    const float* sp = &XL[c * MS + k * 17 + q * 4];
    float4 v;
    v.x = sp[0]; v.y = sp[1]; v.z = sp[2]; v.w = sp[3];
    *(float4*)(dst + (size_t)c * 1048576 + k * 256 + q * 4) = v;
  }
}

// out0 / out2: per (b,s) a 64x64 x 64x16 GEMM; strides parametrized.
__global__ __launch_bounds__(256) void tt_mix_small(
    const float* __restrict__ core, const float* __restrict__ weight,
    float* __restrict__ out, int wrow_off, int sn_in, int ss_in,
    int sn_out, int ss_out) {
  __shared__ float WLp[32 * WLP];    // K-pair-interleaved A
  __shared__ float XLp[32 * 32];     // K-pair-interleaved B (pitch 32: natural stagger)
  int b = blockIdx.x >> 4;
  int s = blockIdx.x & 15;
  int tid = threadIdx.x;

  const float* src = core + (size_t)b * 256 + (size_t)s * ss_in;
  for (int idx = tid; idx < 1024; idx += 256) {
    int i = idx >> 4, n = idx & 15;
    XLp[(i >> 1) * 32 + n * 2 + (i & 1)] = src[(size_t)i * 65536 + (size_t)n * sn_in];
  }
  const float* wsrc = weight + (wrow_off + s) * 64;
  for (int idx = tid; idx < 4096; idx += 256) {
    int i = idx >> 6, c = idx & 63;
    WLp[(i >> 1) * WLP + c * 2 + (i & 1)] = wsrc[i * 3072 + c];
  }
  __syncthreads();

  int wave = tid >> 5;
  if (wave < 4) {                    // uniform branch: EXEC all-ones inside
    int lane = tid & 31, l16 = lane & 15, hi = lane >> 4;
    int mt = wave;
    v2f a_frag[16];
#pragma unroll
    for (int k4 = 0; k4 < 16; ++k4)
      a_frag[k4] = *(const v2f*)&WLp[(k4 * 2 + hi) * WLP + (mt * 16 + l16) * 2];
    const float* bp = &XLp[hi * 32 + l16 * 2];
    v8f ae = {}, ao = {};            // 2 independent chains (K-parity split)
#pragma unroll
    for (int k2 = 0; k2 < 8; ++k2) {
      v2f be = *(const v2f*)&bp[(k2 * 4) * 32];
      v2f bo = *(const v2f*)&bp[(k2 * 4 + 2) * 32];
      ae = __builtin_amdgcn_wmma_f32_16x16x4_f32(false, a_frag[k2 * 2], false, be,
                                                 (short)0, ae, false, false);
      ao = __builtin_amdgcn_wmma_f32_16x16x4_f32(false, a_frag[k2 * 2 + 1], false, bo,
                                                 (short)0, ao, false, false);
    }
    v8f acc = ae + ao;
    float* dst = out + (size_t)b * 256 + (size_t)s * ss_out;
#pragma unroll
    for (int v = 0; v < 8; ++v) {
      int M = mt * 16 + v + hi * 8;
      dst[(size_t)M * 65536 + (size_t)l16 * sn_out] = acc[v];
    }
  }
}

extern "C" void kernel_launch(void* const* d_in, const int* in_sizes, int n_in,
                              void* d_out, int out_size, void* d_ws, size_t ws_size,
                              hipStream_t stream) {
  (void)in_sizes; (void)n_in; (void)d_ws; (void)ws_size; (void)out_size;
  const float* core0  = (const float*)d_in[0];
  const float* core1  = (const float*)d_in[1];
  const float* core2  = (const float*)d_in[2];
  const float* weight = (const float*)d_in[3];
  float* out = (float*)d_out;

  // out0: core0 strides (s=16, j=1); out0 strides (s=16, j=1); w rows 0..15
  tt_mix_small<<<4096, 256, 0, stream>>>(core0, weight, out,
                                         0, 1, 16, 1, 16);

  size_t lds = (size_t)(64 * MS + 32 * XGP + 32 * WLP) * sizeof(float);  // ~156 KB
  tt_out1_fused<<<4096, 256, lds, stream>>>(core1, weight, out + 4194304);

  // out2: core2 strides (k=16, s=1); out2 strides (k=16, s=1); w rows 32..47
  tt_mix_small<<<4096, 256, 0, stream>>>(core2, weight,
                                         out + (4194304 + 67108864),
                                         32, 16, 1, 16, 1);
}